// CRF_69295002353999
// MI455X (gfx1250) — compile-verified
//
#include <hip/hip_runtime.h>
#include <math.h>

typedef __attribute__((ext_vector_type(2))) float    v2f;
typedef __attribute__((ext_vector_type(8))) float    v8f;
typedef __attribute__((ext_vector_type(4))) unsigned u32x4;
typedef __attribute__((ext_vector_type(4))) int      i32x4;
typedef __attribute__((ext_vector_type(8))) int      i32x8;

#define Bsz 256
#define Tsz 512
#define Ksz 128
#define BT  16      // batch rows per workgroup (= WMMA M)
#define PSTR 132    // padded LDS row stride (bank-conflict avoidance)

// ---------------------------------------------------------------------------
// TDM: async 2D tile load (16 rows x 128 f32, row stride T*K) into LDS.
// Descriptor bit layout per CDNA5 ISA sec 8.3-8.5. Issued wave-uniform.
// ---------------------------------------------------------------------------
__device__ __forceinline__ void tdm_load_emit_tile(const float* gsrc, unsigned lds_byte_addr)
{
    unsigned long long ga = (unsigned long long)(uintptr_t)gsrc;
    u32x4 g0;
    g0[0] = 1u;                                     // count=1, user descriptor
    g0[1] = lds_byte_addr;                          // lds_addr
    g0[2] = (unsigned)(ga & 0xffffffffu);           // global_addr[95:64]
    g0[3] = (unsigned)((ga >> 32) & 0x01ffffffu)    // global_addr[120:96]
          | (2u << 30);                             // type=2 ("image")
    i32x8 g1;
    g1[0] = (int)(2u << 16);                        // data_size=4B; no mask/pad/iter
    g1[1] = (int)(128u << 16);                      // tensor_dim0 = 128 elems
    g1[2] = (int)(16u  << 16);                      // tensor_dim1 = 16 rows
    g1[3] = (int)(128u << 16);                      // tile_dim0 = 128
    g1[4] = 16;                                     // tile_dim1 = 16 (tile_dim2 = 0)
    g1[5] = Tsz * Ksz;                              // tensor_dim0_stride = T*K elems
    g1[6] = 0;                                      // stride hi / dim1_stride lo
    g1[7] = 0;                                      // dim1_stride hi
    i32x4 z4 = {0, 0, 0, 0};
#if __has_include(<hip/amd_detail/amd_gfx1250_TDM.h>)
    i32x8 z8 = {0, 0, 0, 0, 0, 0, 0, 0};
    __builtin_amdgcn_tensor_load_to_lds(g0, g1, z4, z4, z8, 0);   // 6-arg toolchain
#else
    __builtin_amdgcn_tensor_load_to_lds(g0, g1, z4, z4, 0);       // 5-arg ROCm 7.2
#endif
}

// ---------------------------------------------------------------------------
// Fused CRF forward scan: Viterbi (max-plus, VALU) + logsumexp (WMMA GEMM),
// emissions streamed by the Tensor Data Mover with double buffering.
// ---------------------------------------------------------------------------
__global__ __launch_bounds__(256) void crf_forward_kernel(
    const float* __restrict__ logits,     // [B][T][K]
    const int*   __restrict__ seq_lens,   // [B]
    const float* __restrict__ trans,      // [K][K]
    unsigned char* __restrict__ bp,       // [B][T][K] backpointers (u8)
    float* __restrict__ vit_alpha,        // [B][K] final viterbi alpha
    float* __restrict__ lse_alpha)        // [B][K] final logsumexp alpha
{
    extern __shared__ unsigned char smem_raw[];
    float* Ecol   = (float*)smem_raw;        // exp(trans) TRANSPOSED: Ecol[c][k], [K][PSTR]
    float* Tr     = Ecol + Ksz * PSTR;       // trans row-major [K][K]
    float* P      = Tr + Ksz * Ksz;          // exp(alpha-m), [BT][PSTR]
    float* La     = P  + BT * PSTR;          // logsumexp alpha, [BT][PSTR]
    float* Va     = La + BT * PSTR;          // viterbi alpha,   [BT][PSTR]
    float* Em     = Va + BT * PSTR;          // emission tiles [2][BT*K] (TDM target)
    float* rowmax = Em + 2 * BT * Ksz;       // [BT]
    float* wmax   = rowmax + BT;             // per-(row,wave) partial maxima [BT][8]
    int*   sl     = (int*)(wmax + BT * 8);   // [BT]

    const int tid  = threadIdx.x;
    const int b0   = blockIdx.x * BT;
    const int lane = tid & 31;
    const int wave = tid >> 5;
    const int lrow = lane & 15;              // 0..15
    const int kh   = lane >> 4;              // 0..1 (wave half)
    const int col  = wave * 16 + lrow;       // this lane's output column (0..127)

    // ---- stage transition matrix: log-domain row-major + exp-domain transposed ----
    for (int idx = tid; idx < Ksz * Ksz; idx += 256) {
        int r = idx >> 7, c = idx & 127;     // trans[r][c]
        float v = trans[idx];
        Tr[idx] = v;
        Ecol[c * PSTR + r] = __expf(v);      // Ecol[c][k]: K-adjacent -> ds_load_b64
    }
    // ---- init both alphas with logits[:, 0, :] ----
    for (int idx = tid; idx < BT * Ksz; idx += 256) {
        int r = idx >> 7, c = idx & 127;
        float v = logits[((size_t)(b0 + r) * Tsz + 0) * Ksz + c];
        La[r * PSTR + c] = v;
        Va[r * PSTR + c] = v;
    }
    if (tid < BT) sl[tid] = seq_lens[b0 + tid];

    // ---- prime the emission pipeline: TDM load of t=1 tile into buffer 0 ----
    if (wave == 0) {
        tdm_load_emit_tile(&logits[((size_t)b0 * Tsz + 1) * Ksz],
                           (unsigned)(uintptr_t)(Em));
    }
    __syncthreads();

    for (int t = 1; t < Tsz; ++t) {
        const int buf = (t - 1) & 1;
        const float* Eb = Em + buf * (BT * Ksz);

        // ---- per-batch-row max of La (logsumexp stabilization) ----
        if (tid < BT) {
            float m;
            if (t == 1) {                    // bootstrap: serial scan of initial La
                m = -INFINITY;
                for (int c = 0; c < Ksz; ++c) m = fmaxf(m, La[tid * PSTR + c]);
            } else {                         // steady state: reduce 8 per-wave partials
                m = wmax[tid * 8];
#pragma unroll
                for (int w = 1; w < 8; ++w) m = fmaxf(m, wmax[tid * 8 + w]);
            }
            rowmax[tid] = m;
        }
        __syncthreads();
        // ---- P = exp(La - rowmax); wave0 also runs the TDM pipeline ----
        for (int idx = tid; idx < BT * Ksz; idx += 256) {
            int r = idx >> 7, c = idx & 127;
            P[r * PSTR + c] = __expf(La[r * PSTR + c] - rowmax[r]);
        }
        if (wave == 0) {
            if (t + 1 < Tsz) {   // kick off next tile, then wait for current one
                tdm_load_emit_tile(&logits[((size_t)b0 * Tsz + (t + 1)) * Ksz],
                                   (unsigned)(uintptr_t)(Em + (buf ^ 1) * (BT * Ksz)));
                __builtin_amdgcn_s_wait_tensorcnt(1);
            } else {
                __builtin_amdgcn_s_wait_tensorcnt(0);
            }
        }
        __syncthreads();   // releases emission tile t + P to all waves

        // ---- WMMA: acc[16x16] = P(16x128) x E(128 x this wave's 16 cols) ----
        // 4 independent accumulator chains (8 WMMAs each) to keep the matrix
        // pipe filled instead of one 32-deep serial C-dependency chain.
        v8f acc0 = {}, acc1 = {}, acc2 = {}, acc3 = {};
#pragma unroll
        for (int kb = 0; kb < 32; kb += 4) {
            const int o0 = kb + 2 * kh;
            v2f a0 = *(const v2f*)&P[lrow * PSTR + o0];
            v2f b0v = *(const v2f*)&Ecol[col * PSTR + o0];
            acc0 = __builtin_amdgcn_wmma_f32_16x16x4_f32(false, a0, false, b0v,
                                                         (short)0, acc0, false, false);
            v2f a1 = *(const v2f*)&P[lrow * PSTR + o0 + 32];
            v2f b1v = *(const v2f*)&Ecol[col * PSTR + o0 + 32];
            acc1 = __builtin_amdgcn_wmma_f32_16x16x4_f32(false, a1, false, b1v,
                                                         (short)0, acc1, false, false);
            v2f a2 = *(const v2f*)&P[lrow * PSTR + o0 + 64];
            v2f b2v = *(const v2f*)&Ecol[col * PSTR + o0 + 64];
            acc2 = __builtin_amdgcn_wmma_f32_16x16x4_f32(false, a2, false, b2v,
                                                         (short)0, acc2, false, false);
            v2f a3 = *(const v2f*)&P[lrow * PSTR + o0 + 96];
            v2f b3v = *(const v2f*)&Ecol[col * PSTR + o0 + 96];
            acc3 = __builtin_amdgcn_wmma_f32_16x16x4_f32(false, a3, false, b3v,
                                                         (short)0, acc3, false, false);
        }
        v8f acc = (acc0 + acc1) + (acc2 + acc3);

        // ---- phase 1: read emissions/keeps + Viterbi max-plus (VALU) ----
        float emitv[8], keepL[8], keepV[8], vbest[8];
        int   varg[8];
#pragma unroll
        for (int r = 0; r < 8; ++r) {
            int br = r + 8 * kh;
            emitv[r] = Eb[br * Ksz + col];
            keepL[r] = La[br * PSTR + col];
            keepV[r] = Va[br * PSTR + col];
            vbest[r] = -INFINITY;
            varg[r]  = 0;
        }
        // K-predecessor loop unrolled x2; Va fetched as aligned ds_load_b64.
        for (int i = 0; i < Ksz; i += 2) {
            float trv0 = Tr[(i + 0) * Ksz + col];   // same word for lanes l and l+16
            float trv1 = Tr[(i + 1) * Ksz + col];
#pragma unroll
            for (int r = 0; r < 8; ++r) {
                int br = r + 8 * kh;
                v2f va2 = *(const v2f*)&Va[br * PSTR + i];
                float s0 = va2.x + trv0;
                bool  g0 = s0 > vbest[r];           // strict > keeps first argmax
                varg[r]  = g0 ? i : varg[r];
                vbest[r] = g0 ? s0 : vbest[r];
                float s1 = va2.y + trv1;
                bool  g1 = s1 > vbest[r];
                varg[r]  = g1 ? (i + 1) : varg[r];
                vbest[r] = g1 ? s1 : vbest[r];
            }
        }
        __syncthreads();   // all reads of La/Va for step t complete

        // ---- phase 2: masked state update + backpointers; also compute the
        //      per-(row,wave) La maxima via half-wave shuffle reduction so the
        //      next step's rowmax is an 8-element reduce instead of a 128 scan.
        float newLa[8];
#pragma unroll
        for (int r = 0; r < 8; ++r) {
            int  br = r + 8 * kh;
            bool on = (t < sl[br]);
            float lnew = rowmax[br] + __logf(acc[r]) + emitv[r];
            float vnew = vbest[r] + emitv[r];
            newLa[r] = on ? lnew : keepL[r];
            La[br * PSTR + col] = newLa[r];
            Va[br * PSTR + col] = on ? vnew : keepV[r];
            bp[((size_t)(b0 + br) * Tsz + t) * Ksz + col] =
                (unsigned char)(on ? varg[r] : col);   // identity bp when masked
        }
#pragma unroll
        for (int r = 0; r < 8; ++r) {
            float m = newLa[r];                    // row br, this wave's 16 columns
            m = fmaxf(m, __shfl_xor(m, 1, 32));
            m = fmaxf(m, __shfl_xor(m, 2, 32));
            m = fmaxf(m, __shfl_xor(m, 4, 32));
            m = fmaxf(m, __shfl_xor(m, 8, 32));
            if (lrow == 0) wmax[(r + 8 * kh) * 8 + wave] = m;
        }
        __syncthreads();
    }

    // ---- export final alphas for backtrace / loss kernels ----
    for (int idx = tid; idx < BT * Ksz; idx += 256) {
        int r = idx >> 7, c = idx & 127;
        vit_alpha[(size_t)(b0 + r) * Ksz + c] = Va[r * PSTR + c];
        lse_alpha[(size_t)(b0 + r) * Ksz + c] = La[r * PSTR + c];
    }
}

// ---------------------------------------------------------------------------
// Per-batch Viterbi backtrace.
// ---------------------------------------------------------------------------
__global__ __launch_bounds__(256) void crf_backtrace_kernel(
    const unsigned char* __restrict__ bp,
    const float* __restrict__ vit_alpha,
    float* __restrict__ out_pred)            // d_out + 1, [B][T] as float
{
    int b = blockIdx.x * blockDim.x + threadIdx.x;
    if (b >= Bsz) return;
    float best = vit_alpha[(size_t)b * Ksz];
    int   tag  = 0;
    for (int j = 1; j < Ksz; ++j) {
        float v = vit_alpha[(size_t)b * Ksz + j];
        if (v > best) { best = v; tag = j; }
    }
    out_pred[(size_t)b * Tsz + (Tsz - 1)] = (float)tag;
    for (int t = Tsz - 1; t >= 1; --t) {
        tag = bp[((size_t)b * Tsz + t) * Ksz + tag];
        out_pred[(size_t)b * Tsz + (t - 1)] = (float)tag;
    }
}

// ---------------------------------------------------------------------------
// Loss: per-batch -(score - logZ), deterministically tree-reduced in one block.
// ---------------------------------------------------------------------------
__global__ __launch_bounds__(256) void crf_loss_kernel(
    const float* __restrict__ logits,
    const int*   __restrict__ labels,
    const int*   __restrict__ seq_lens,
    const float* __restrict__ trans,
    const float* __restrict__ lse_alpha,
    float* __restrict__ out_loss)            // d_out[0]
{
    __shared__ float red[256];
    int b = threadIdx.x;
    int L = seq_lens[b];
    int prev = labels[(size_t)b * Tsz];
    float unary  = logits[((size_t)b * Tsz) * Ksz + prev];   // t=0 always < L (L>=1)
    float binary = 0.f;
    for (int t = 1; t < Tsz; ++t) {
        int   cur = labels[(size_t)b * Tsz + t];
        float mk  = (t < L) ? 1.f : 0.f;
        unary  += logits[((size_t)b * Tsz + t) * Ksz + cur] * mk;
        binary += trans[prev * Ksz + cur] * mk;
        prev = cur;
    }
    float m = -INFINITY;
    for (int j = 0; j < Ksz; ++j) m = fmaxf(m, lse_alpha[(size_t)b * Ksz + j]);
    float s = 0.f;
    for (int j = 0; j < Ksz; ++j) s += __expf(lse_alpha[(size_t)b * Ksz + j] - m);
    float logZ = m + __logf(s);
    red[b] = logZ - (unary + binary);        // -log_likelihood[b]
    __syncthreads();
    for (int off = 128; off > 0; off >>= 1) {
        if (b < off) red[b] += red[b + off];
        __syncthreads();
    }
    if (b == 0) out_loss[0] = red[0];
}

// ---------------------------------------------------------------------------
extern "C" void kernel_launch(void* const* d_in, const int* in_sizes, int n_in,
                              void* d_out, int out_size, void* d_ws, size_t ws_size,
                              hipStream_t stream)
{
    const float* logits   = (const float*)d_in[0];
    const int*   labels   = (const int*)d_in[1];
    const int*   seq_lens = (const int*)d_in[2];
    const float* trans    = (const float*)d_in[3];
    float* out = (float*)d_out;              // [0]=loss, [1..]=pred_ids (as float)

    unsigned char* bp = (unsigned char*)d_ws;                       // 16 MB
    float* vit_alpha  = (float*)(bp + (size_t)Bsz * Tsz * Ksz);     // B*K
    float* lse_alpha  = vit_alpha + (size_t)Bsz * Ksz;              // B*K

    size_t shmem = (size_t)(Ksz * PSTR            // Ecol
                          + Ksz * Ksz             // Tr
                          + 3 * BT * PSTR         // P, La, Va
                          + 2 * BT * Ksz          // emission double buffer
                          + BT                    // rowmax
                          + BT * 8) * sizeof(float)  // wmax partials
                 + BT * sizeof(int);              // ~172 KB of the WGP's 320 KB LDS

    crf_forward_kernel<<<Bsz / BT, 256, shmem, stream>>>(
        logits, seq_lens, trans, bp, vit_alpha, lse_alpha);
    crf_backtrace_kernel<<<1, 256, 0, stream>>>(bp, vit_alpha, out + 1);
    crf_loss_kernel<<<1, 256, 0, stream>>>(
        logits, labels, seq_lens, trans, lse_alpha, out);
}